// SCAT_Red_50714973831801
// MI455X (gfx1250) — compile-verified
//
#include <hip/hip_runtime.h>
#include <hip/hip_bf16.h>

// ---------------------------------------------------------------------------
// SCAT GNN layer for MI455X (gfx1250, wave32).
//   out[N,640] = concat(s0..s4), s_i = H_i @ W_i + b_i
//   H0..H2 = A_tilde^k @ X  (SpMM chain, atomic scatter)
//   H3 = P^o1 X - P^{2o1} X,  H4 = P^o2 X - P^{2o2} X,  P = 0.5(adj D^-1 + I)
// Dense GEMMs use V_WMMA_F32_16X16X4_F32 (f32 in/out, 16x16 tiles, K-step 4).
// ---------------------------------------------------------------------------

#define F_DIM 128
#define OUT_STRIDE 640

typedef __attribute__((ext_vector_type(2))) float v2f;
typedef __attribute__((ext_vector_type(8))) float v8f;

// ---------------------------------------------------------------------------
// SpMM scatter: y[r,:] += scale * v * x[c,:]  (one wave per edge, 4 floats/lane)
// ---------------------------------------------------------------------------
__global__ __launch_bounds__(256) void spmm_atomic_kernel(
    const int* __restrict__ rows, const int* __restrict__ cols,
    const float* __restrict__ vals, const float* __restrict__ x,
    float* __restrict__ y, int E, float scale) {
  int wave = blockIdx.x * 8 + (threadIdx.x >> 5);
  if (wave >= E) return;
  int lane = threadIdx.x & 31;
  int r = rows[wave];
  int c = cols[wave];
  float v = vals[wave] * scale;
  float4 xv = ((const float4*)(x + (size_t)c * F_DIM))[lane];
  float* yp = y + (size_t)r * F_DIM + lane * 4;
  unsafeAtomicAdd(yp + 0, v * xv.x);
  unsafeAtomicAdd(yp + 1, v * xv.y);
  unsafeAtomicAdd(yp + 2, v * xv.z);
  unsafeAtomicAdd(yp + 3, v * xv.w);
}

// deg[c] += vals[e]
__global__ __launch_bounds__(256) void degree_kernel(
    const int* __restrict__ cols, const float* __restrict__ vals,
    float* __restrict__ deg, int E) {
  int e = blockIdx.x * 256 + threadIdx.x;
  if (e >= E) return;
  unsafeAtomicAdd(&deg[cols[e]], vals[e]);
}

// nvals[e] = vals[e] / deg[cols[e]]  (0 when deg == 0)
__global__ __launch_bounds__(256) void normalize_kernel(
    const int* __restrict__ cols, const float* __restrict__ vals,
    const float* __restrict__ deg, float* __restrict__ nvals, int E) {
  int e = blockIdx.x * 256 + threadIdx.x;
  if (e >= E) return;
  float d = deg[cols[e]];
  nvals[e] = vals[e] * (d > 0.0f ? 1.0f / d : 0.0f);
}

// y = 0.5 * x  (float4 vectorized; n4 = count of float4)
__global__ __launch_bounds__(256) void half_copy_kernel(
    const float4* __restrict__ x, float4* __restrict__ y, int n4) {
  int i = blockIdx.x * 256 + threadIdx.x;
  if (i >= n4) return;
  float4 v = x[i];
  v.x *= 0.5f; v.y *= 0.5f; v.z *= 0.5f; v.w *= 0.5f;
  y[i] = v;
}

// ---------------------------------------------------------------------------
// WMMA GEMM core: one block = 16 rows x 128 cols of C. 8 waves, each wave a
// 16x16 tile. A tile (16x128 f32) staged in LDS; K-loop of 32 x wmma_16x16x4.
// A frag (16x4):  lane<16: rows, VGPR0=K, VGPR1=K+1 ; lanes 16..31: K+2,K+3.
// B frag (4x16):  lane%16 = col, K = kk + 2*(lane/16) + vgpr.
// C/D (16x16):    VGPR j -> row = j + 8*(lane/16), col = lane%16.
// ---------------------------------------------------------------------------
__device__ inline void gemm_tile_core(const float* __restrict__ ldsA,
                                      const float* __restrict__ W,
                                      const float* __restrict__ bias,
                                      float* __restrict__ out, int rowBase,
                                      int N, int outColBase) {
  const int wv = threadIdx.x >> 5;
  const int lane = threadIdx.x & 31;
  const int lh = lane >> 4;   // lane half
  const int lr = lane & 15;   // row (A) / col (B,C)
  const int colBase = wv * 16;

  v8f c = {};
#pragma unroll
  for (int kk = 0; kk < F_DIM; kk += 4) {
    const int k0 = kk + 2 * lh;
    v2f a, b;
    a.x = ldsA[lr * F_DIM + k0];
    a.y = ldsA[lr * F_DIM + k0 + 1];
    b.x = W[(size_t)k0 * F_DIM + colBase + lr];
    b.y = W[(size_t)(k0 + 1) * F_DIM + colBase + lr];
    c = __builtin_amdgcn_wmma_f32_16x16x4_f32(
        /*neg_a=*/false, a, /*neg_b=*/false, b,
        /*c_mod=*/(short)0, c, /*reuse_a=*/false, /*reuse_b=*/false);
  }

  const float bv = bias[colBase + lr];
#pragma unroll
  for (int j = 0; j < 8; ++j) {
    const int row = rowBase + j + 8 * lh;
    if (row < N)
      out[(size_t)row * OUT_STRIDE + outColBase + colBase + lr] = c[j] + bv;
  }
}

// C[:, outColBase:+128] = (A - A2?) @ W + b
__global__ __launch_bounds__(256) void gemm_bias_kernel(
    const float* __restrict__ A, const float* __restrict__ A2,
    const float* __restrict__ W, const float* __restrict__ bias,
    float* __restrict__ out, int N, int outColBase) {
  __shared__ float ldsA[16 * F_DIM];
  const int rowBase = blockIdx.x * 16;
  for (int i = threadIdx.x; i < 16 * (F_DIM / 4); i += 256) {
    const int row = i >> 5;   // 32 float4 per row
    const int q = i & 31;
    const int g = rowBase + row;
    float4 v = make_float4(0.f, 0.f, 0.f, 0.f);
    if (g < N) {
      v = ((const float4*)(A + (size_t)g * F_DIM))[q];
      if (A2 != nullptr) {
        float4 u = ((const float4*)(A2 + (size_t)g * F_DIM))[q];
        v.x -= u.x; v.y -= u.y; v.z -= u.z; v.w -= u.w;
      }
    }
    ((float4*)ldsA)[i] = v;
  }
  __syncthreads();
  gemm_tile_core(ldsA, W, bias, out, rowBase, N, outColBase);
}

// Same GEMM but A = powers[o] - powers[2o], o read from device scalar.
struct PowPtrs { const float* p[5]; };

__global__ __launch_bounds__(256) void gemm_powdiff_kernel(
    PowPtrs pw, const int* __restrict__ order, const float* __restrict__ W,
    const float* __restrict__ bias, float* __restrict__ out, int N,
    int outColBase) {
  int o = order[0];
  o = o < 0 ? 0 : (o > 2 ? 2 : o);
  const float* A = pw.p[o];
  const float* B = pw.p[2 * o];

  __shared__ float ldsA[16 * F_DIM];
  const int rowBase = blockIdx.x * 16;
  for (int i = threadIdx.x; i < 16 * (F_DIM / 4); i += 256) {
    const int row = i >> 5;
    const int q = i & 31;
    const int g = rowBase + row;
    float4 v = make_float4(0.f, 0.f, 0.f, 0.f);
    if (g < N) {
      float4 a4 = ((const float4*)(A + (size_t)g * F_DIM))[q];
      float4 b4 = ((const float4*)(B + (size_t)g * F_DIM))[q];
      v.x = a4.x - b4.x; v.y = a4.y - b4.y;
      v.z = a4.z - b4.z; v.w = a4.w - b4.w;
    }
    ((float4*)ldsA)[i] = v;
  }
  __syncthreads();
  gemm_tile_core(ldsA, W, bias, out, rowBase, N, outColBase);
}

// ---------------------------------------------------------------------------
extern "C" void kernel_launch(void* const* d_in, const int* in_sizes, int n_in,
                              void* d_out, int out_size, void* d_ws,
                              size_t ws_size, hipStream_t stream) {
  // setup_inputs() dict order:
  const float* features = (const float*)d_in[0];
  const int*   at_rows  = (const int*)d_in[1];
  const int*   at_cols  = (const int*)d_in[2];
  const float* at_vals  = (const float*)d_in[3];
  const int*   adj_rows = (const int*)d_in[4];
  const int*   adj_cols = (const int*)d_in[5];
  const float* adj_vals = (const float*)d_in[6];
  const float* Wm[5], *bm[5];
  for (int i = 0; i < 5; ++i) {
    Wm[i] = (const float*)d_in[7 + 2 * i];
    bm[i] = (const float*)d_in[8 + 2 * i];
  }
  const int* order1 = (const int*)d_in[17];
  const int* order2 = (const int*)d_in[18];

  const int N = in_sizes[0] / F_DIM;
  const int E = in_sizes[1];
  float* out = (float*)d_out;

  // Workspace carve-up.
  const size_t bufElems = (size_t)N * F_DIM;
  const size_t bufBytes = bufElems * sizeof(float);
  char* w = (char*)d_ws;
  float* buf0 = (float*)w; w += bufBytes;   // h1 / h3
  float* buf1 = (float*)w; w += bufBytes;   // h2
  float* p1   = (float*)w; w += bufBytes;
  float* p2   = (float*)w; w += bufBytes;
  float* p3   = (float*)w; w += bufBytes;
  float* p4   = (float*)w; w += bufBytes;
  float* deg  = (float*)w; w += (size_t)N * sizeof(float);
  float* nvals = (float*)w; w += (size_t)E * sizeof(float);

  const dim3 blk(256);
  const int spmmBlocks = (E + 7) / 8;           // 8 edges (waves) per block
  const int edgeBlocks = (E + 255) / 256;
  const int gemmBlocks = (N + 15) / 16;
  const int n4 = (int)(bufElems / 4);
  const int copyBlocks = (n4 + 255) / 256;

  // ---- A_tilde hop chain + s0..s2 --------------------------------------
  hipMemsetAsync(buf0, 0, bufBytes, stream);
  spmm_atomic_kernel<<<spmmBlocks, blk, 0, stream>>>(at_rows, at_cols, at_vals,
                                                     features, buf0, E, 1.0f);
  gemm_bias_kernel<<<gemmBlocks, blk, 0, stream>>>(buf0, nullptr, Wm[0], bm[0],
                                                   out, N, 0 * F_DIM);

  hipMemsetAsync(buf1, 0, bufBytes, stream);
  spmm_atomic_kernel<<<spmmBlocks, blk, 0, stream>>>(at_rows, at_cols, at_vals,
                                                     buf0, buf1, E, 1.0f);
  gemm_bias_kernel<<<gemmBlocks, blk, 0, stream>>>(buf1, nullptr, Wm[1], bm[1],
                                                   out, N, 1 * F_DIM);

  hipMemsetAsync(buf0, 0, bufBytes, stream);
  spmm_atomic_kernel<<<spmmBlocks, blk, 0, stream>>>(at_rows, at_cols, at_vals,
                                                     buf1, buf0, E, 1.0f);
  gemm_bias_kernel<<<gemmBlocks, blk, 0, stream>>>(buf0, nullptr, Wm[2], bm[2],
                                                   out, N, 2 * F_DIM);

  // ---- column-degree normalization -------------------------------------
  hipMemsetAsync(deg, 0, (size_t)N * sizeof(float), stream);
  degree_kernel<<<edgeBlocks, blk, 0, stream>>>(adj_cols, adj_vals, deg, E);
  normalize_kernel<<<edgeBlocks, blk, 0, stream>>>(adj_cols, adj_vals, deg,
                                                   nvals, E);

  // ---- lazy-walk powers p_k = 0.5*(adjnorm @ p_{k-1} + p_{k-1}) --------
  const float* src = features;
  float* pows[4] = {p1, p2, p3, p4};
  for (int k = 0; k < 4; ++k) {
    half_copy_kernel<<<copyBlocks, blk, 0, stream>>>((const float4*)src,
                                                     (float4*)pows[k], n4);
    spmm_atomic_kernel<<<spmmBlocks, blk, 0, stream>>>(
        adj_rows, adj_cols, nvals, src, pows[k], E, 0.5f);
    src = pows[k];
  }

  // ---- s3 / s4: (P^o - P^{2o}) X @ W + b, order from device scalars ----
  PowPtrs pw;
  pw.p[0] = features; pw.p[1] = p1; pw.p[2] = p2; pw.p[3] = p3; pw.p[4] = p4;
  gemm_powdiff_kernel<<<gemmBlocks, blk, 0, stream>>>(pw, order1, Wm[3], bm[3],
                                                      out, N, 3 * F_DIM);
  gemm_powdiff_kernel<<<gemmBlocks, blk, 0, stream>>>(pw, order2, Wm[4], bm[4],
                                                      out, N, 4 * F_DIM);
}